// SoftSort_p2_8778913153217
// MI455X (gfx1250) — compile-verified
//
#include <hip/hip_runtime.h>
#include <math.h>
#include <stdint.h>

typedef __attribute__((ext_vector_type(2))) float v2f;
typedef __attribute__((ext_vector_type(8))) float v8f;

constexpr int NN      = 2048;       // row length
constexpr int NT      = 256;        // threads per block (8 waves, wave32)
constexpr int NWAVES  = NT / 32;
constexpr int CHUNKS  = 8;          // i-chunks per row
constexpr int IPB     = NN / CHUNKS;     // 256 i's per block
constexpr int TILES   = IPB / 16;        // 16 i-tiles of 16 per block
constexpr float INV_TAU = 1.0e4f;        // 1/tau

typedef __attribute__((address_space(1))) int* gas_int_ptr;
typedef __attribute__((address_space(3))) void* las_void_ptr;
typedef __attribute__((address_space(3))) int* las_int_ptr;

__global__ __launch_bounds__(NT)
void softsort_kernel(const float* __restrict__ scores, float* __restrict__ out)
{
    __shared__ float sBuf[NN];                 // original row
    __shared__ float tBuf[NN];                 // sorted (descending) row
    __shared__ float red[NWAVES][16][2];       // per-wave num/den scatter

    const int b     = blockIdx.y;
    const int chunk = blockIdx.x;
    const int tid   = threadIdx.x;
    const float* row = scores + (size_t)b * NN;

    // ---- Stage 1: row -> LDS (async global->LDS path when available) ----
#if __has_builtin(__builtin_amdgcn_global_load_async_to_lds_b32)
    for (int i = tid; i < NN; i += NT) {
        __builtin_amdgcn_global_load_async_to_lds_b32(
            (gas_int_ptr)(uintptr_t)(row + i),
            (las_int_ptr)(las_void_ptr)(sBuf + i),
            0, 0);
    }
    asm volatile("s_wait_asynccnt 0" ::: "memory");
#else
    for (int i = tid; i < NN; i += NT) sBuf[i] = row[i];
#endif
    __syncthreads();
    for (int i = tid; i < NN; i += NT) tBuf[i] = sBuf[i];

    // ---- Stage 2: bitonic sort, descending, in LDS ----
    for (int k = 2; k <= NN; k <<= 1) {
        for (int j = k >> 1; j > 0; j >>= 1) {
            __syncthreads();
            for (int i = tid; i < NN; i += NT) {
                int ix = i ^ j;
                if (ix > i) {
                    float a = tBuf[i], c = tBuf[ix];
                    bool descBlk = ((i & k) == 0);
                    if (descBlk ? (a < c) : (a > c)) { tBuf[i] = c; tBuf[ix] = a; }
                }
            }
        }
    }
    __syncthreads();

    // ---- Stage 3: fused exp + dual reduction via V_WMMA_F32_16X16X4_F32 ----
    const int wave  = tid >> 5;
    const int lane  = tid & 31;
    const int m     = lane & 15;     // A: row M (both lane halves map M=0..15)
    const int khalf = lane >> 4;     // A: VGPR0 holds K = 2*khalf, VGPR1 K = 2*khalf+1
    const bool n0   = (m == 0);      // B column 0 -> s_j (numerator)
    const float c1  = (m == 1) ? 1.0f : 0.0f;   // B column 1 -> 1.0 (denominator)

    for (int tile = wave; tile < TILES; tile += NWAVES) {
        const int i0  = chunk * IPB + tile * 16;
        const float ti = tBuf[i0 + m];

        v8f acc0 = {}, acc1 = {};
        for (int kb = 0; kb < NN; kb += 8) {
            // --- j-slice [kb, kb+4) ---
            {
                const float2 sp = *(const float2*)&sBuf[kb + khalf * 2];
                float d0 = sp.x - ti, d1 = sp.y - ti;
                v2f a, bb;
                a.x = __expf(-(d0 * d0) * INV_TAU);
                a.y = __expf(-(d1 * d1) * INV_TAU);
                bb.x = n0 ? sp.x : c1;
                bb.y = n0 ? sp.y : c1;
                acc0 = __builtin_amdgcn_wmma_f32_16x16x4_f32(
                    false, a, false, bb, (short)0, acc0, false, false);
            }
            // --- j-slice [kb+4, kb+8) (second accumulator hides WMMA latency) ---
            {
                const float2 sp = *(const float2*)&sBuf[kb + 4 + khalf * 2];
                float d0 = sp.x - ti, d1 = sp.y - ti;
                v2f a, bb;
                a.x = __expf(-(d0 * d0) * INV_TAU);
                a.y = __expf(-(d1 * d1) * INV_TAU);
                bb.x = n0 ? sp.x : c1;
                bb.y = n0 ? sp.y : c1;
                acc1 = __builtin_amdgcn_wmma_f32_16x16x4_f32(
                    false, a, false, bb, (short)0, acc1, false, false);
            }
        }
        v8f acc = acc0 + acc1;

        // D layout: VGPR r -> (M=r, N=lane) lanes 0-15 ; (M=r+8, N=lane-16) lanes 16-31.
        // Column N=0 = numerator, N=1 = denominator.
        if (lane == 0)  { for (int r = 0; r < 8; ++r) red[wave][r][0]     = acc[r]; }
        if (lane == 1)  { for (int r = 0; r < 8; ++r) red[wave][r][1]     = acc[r]; }
        if (lane == 16) { for (int r = 0; r < 8; ++r) red[wave][r + 8][0] = acc[r]; }
        if (lane == 17) { for (int r = 0; r < 8; ++r) red[wave][r + 8][1] = acc[r]; }
        asm volatile("" ::: "memory");   // wave-local LDS RAW; backend inserts s_wait_dscnt
        if (lane < 16) {
            out[(size_t)b * NN + i0 + lane] = red[wave][lane][0] / red[wave][lane][1];
        }
    }
}

extern "C" void kernel_launch(void* const* d_in, const int* in_sizes, int n_in,
                              void* d_out, int out_size, void* d_ws, size_t ws_size,
                              hipStream_t stream)
{
    const float* scores = (const float*)d_in[0];
    float* out = (float*)d_out;
    const int Bn = in_sizes[0] / NN;          // 64 rows
    dim3 grid(CHUNKS, Bn, 1);
    softsort_kernel<<<grid, NT, 0, stream>>>(scores, out);
}